// LightGCN_11416023072988
// MI455X (gfx1250) — compile-verified
//
#include <hip/hip_runtime.h>
#include <hip/hip_bf16.h>

#define NUSERS 100000
#define NITEMS 50000
#define NNODES 150000           // NUSERS + NITEMS
#define EMB    64
#define NEDGES 2400000
#define NLAYERS 3
#define TILE   256              // edges per staged tile (2.4M % 256 == 0 -> 9375 tiles)
#define SCATTER_BLOCKS 2048     // ~4.6 tiles/block -> pipeline has work to overlap

#ifndef __has_builtin
#define __has_builtin(x) 0
#endif

// CDNA5 async global->LDS copy path (ASYNCcnt-tracked).
#if defined(__HIP_DEVICE_COMPILE__) && \
    __has_builtin(__builtin_amdgcn_global_load_async_to_lds_b32) && \
    __has_builtin(__builtin_amdgcn_s_wait_asynccnt)
#define USE_ASYNC_LDS 1
#else
#define USE_ASYNC_LDS 0
#endif

// ---------------- degree / norm precompute ----------------

__global__ __launch_bounds__(256) void k_fill1(float* deg) {
    int i = blockIdx.x * blockDim.x + threadIdx.x;
    if (i < NNODES) deg[i] = 1.0f;              // self-loop contributes 1 to every node
}

__global__ __launch_bounds__(256) void k_deg_accum(const int* __restrict__ col,
                                                   float* __restrict__ deg) {
    int e = blockIdx.x * blockDim.x + threadIdx.x;   // exactly NEDGES threads
    atomicAdd(&deg[col[e]], 1.0f);
}

__global__ __launch_bounds__(256) void k_dinv(float* dinv) {
    int i = blockIdx.x * blockDim.x + threadIdx.x;
    if (i < NNODES) {
        float d = dinv[i];                       // deg >= 1 always (self-loops)
        dinv[i] = 1.0f / sqrtf(d);
    }
}

__global__ __launch_bounds__(256) void k_edge_norm(const int* __restrict__ row,
                                                   const int* __restrict__ col,
                                                   const float* __restrict__ dinv,
                                                   float* __restrict__ enorm) {
    int e = blockIdx.x * blockDim.x + threadIdx.x;   // exactly NEDGES threads
    enorm[e] = dinv[row[e]] * dinv[col[e]];
}

// ---------------- embedding init: x_cur = concat(user,item); final = w0 * x ----------------

__global__ __launch_bounds__(256) void k_init(const float4* __restrict__ user,
                                              const float4* __restrict__ item,
                                              const float* __restrict__ lw,
                                              float4* __restrict__ x_cur,
                                              float4* __restrict__ out) {
    int i = blockIdx.x * blockDim.x + threadIdx.x;   // exactly NNODES*16 threads (float4)
    float w0 = lw[0];
    int node = i >> 4;
    float4 v = (node < NUSERS) ? user[i] : item[i - NUSERS * 16];
    x_cur[i] = v;
    out[i] = make_float4(w0 * v.x, w0 * v.y, w0 * v.z, w0 * v.w);
}

// ---------------- self-loop term: x_next = dinv[node]^2 * x_cur ----------------

__global__ __launch_bounds__(256) void k_selfloop(const float4* __restrict__ x_cur,
                                                  const float* __restrict__ dinv,
                                                  float4* __restrict__ x_next) {
    int i = blockIdx.x * blockDim.x + threadIdx.x;   // NNODES*16
    float s = dinv[i >> 4];
    s *= s;
    float4 v = x_cur[i];
    x_next[i] = make_float4(s * v.x, s * v.y, s * v.z, s * v.w);
}

// ---------------- edge scatter: x_next[col] += norm * x_cur[row] ----------------
// 16 threads per edge (float4 chunk each). Edge tiles (row/col/norm) are
// double-buffered in LDS: while the atomic inner loop processes tile i from
// buffer `buf`, the CDNA5 async DMA streams tile i+1 into buffer `buf^1`.
// Async copies complete in order, so with 6 outstanding ops
// `s_wait_asynccnt 3` releases exactly the current tile's 3 copies.

__global__ __launch_bounds__(TILE) void k_scatter(const int* __restrict__ row,
                                                  const int* __restrict__ col,
                                                  const float* __restrict__ enorm,
                                                  const float4* __restrict__ xv,
                                                  float* __restrict__ xo) {
    __shared__ int   s_row[2][TILE];
    __shared__ int   s_col[2][TILE];
    __shared__ float s_nrm[2][TILE];

    const int t      = threadIdx.x;            // blockDim.x == TILE == 256
    const int ntiles = NEDGES / TILE;          // 9375

    auto stage = [&](int b, int tl) {
        const int base = tl * TILE;
#if USE_ASYNC_LDS
        __builtin_amdgcn_global_load_async_to_lds_b32(
            (int*)(row + base + t),   (int*)&s_row[b][t], 0, 0);
        __builtin_amdgcn_global_load_async_to_lds_b32(
            (int*)(col + base + t),   (int*)&s_col[b][t], 0, 0);
        __builtin_amdgcn_global_load_async_to_lds_b32(
            (int*)(enorm + base + t), (int*)&s_nrm[b][t], 0, 0);
#else
        s_row[b][t] = row[base + t];
        s_col[b][t] = col[base + t];
        s_nrm[b][t] = enorm[base + t];
#endif
    };

    int tile = blockIdx.x;
    if (tile >= ntiles) return;

    stage(0, tile);                            // prologue: fill buffer 0
    int buf = 0;
    for (; tile < ntiles; tile += gridDim.x) {
        const int  next     = tile + gridDim.x;
        const bool has_next = next < ntiles;
        if (has_next) stage(buf ^ 1, next);    // overlap DMA of next tile
#if USE_ASYNC_LDS
        if (has_next) __builtin_amdgcn_s_wait_asynccnt(3);  // current tile's 3 done
        else          __builtin_amdgcn_s_wait_asynccnt(0);
#endif
        __syncthreads();                       // current buffer visible to all waves

        // prefetch the gather row this thread's edge will touch
        __builtin_prefetch((const void*)&xv[(long)s_row[buf][t] * 16], 0, 3);

        const int sub = t >> 4;                // which of 16 edges in this sweep
        const int c   = t & 15;                // float4 chunk within the 64-dim row
#pragma unroll
        for (int j = 0; j < 16; ++j) {
            const int   e  = j * 16 + sub;
            const int   r  = s_row[buf][e];
            const int   cl = s_col[buf][e];
            const float w  = s_nrm[buf][e];
            float4 v = xv[(long)r * 16 + c];
            float* dst = xo + (long)cl * EMB + c * 4;
            atomicAdd(dst + 0, w * v.x);
            atomicAdd(dst + 1, w * v.y);
            atomicAdd(dst + 2, w * v.z);
            atomicAdd(dst + 3, w * v.w);
        }
        __syncthreads();                       // all reads of `buf` done before restage
        buf ^= 1;
    }
}

// ---------------- final += w[l] * x ----------------

__global__ __launch_bounds__(256) void k_axpy(float4* __restrict__ out,
                                              const float4* __restrict__ x,
                                              const float* __restrict__ lw, int li) {
    int i = blockIdx.x * blockDim.x + threadIdx.x;   // NNODES*16
    float w = lw[li];
    float4 a = out[i], b = x[i];
    out[i] = make_float4(a.x + w * b.x, a.y + w * b.y, a.z + w * b.z, a.w + w * b.w);
}

// ---------------- row-wise L2 normalize (one wave32 per 64-float row) ----------------

__global__ __launch_bounds__(256) void k_l2norm(float* __restrict__ out) {
    int r    = blockIdx.x * 8 + (threadIdx.x >> 5);   // 8 waves/block -> 8 rows
    int lane = threadIdx.x & 31;
    if (r >= NNODES) return;
    float2* p = (float2*)(out + (size_t)r * EMB) + lane;  // 32 float2 per row
    float2 v = *p;
    float s = v.x * v.x + v.y * v.y;
#pragma unroll
    for (int m = 16; m >= 1; m >>= 1) s += __shfl_xor(s, m, 32);
    float scale = 1.0f / fmaxf(sqrtf(s), 1e-12f);
    v.x *= scale; v.y *= scale;
    *p = v;
}

// ---------------- host launcher ----------------

extern "C" void kernel_launch(void* const* d_in, const int* in_sizes, int n_in,
                              void* d_out, int out_size, void* d_ws, size_t ws_size,
                              hipStream_t stream) {
    const int*   edge = (const int*)d_in[0];          // [2, NEDGES]
    const float* user = (const float*)d_in[1];        // [NUSERS, 64]
    const float* item = (const float*)d_in[2];        // [NITEMS, 64]
    const float* lw   = (const float*)d_in[3];        // [4]
    float* out = (float*)d_out;                       // [NNODES, 64]

    const int* row = edge;
    const int* col = edge + NEDGES;

    // carve workspace (256B aligned)
    char* ws = (char*)d_ws;
    auto carve = [&](size_t bytes) {
        void* p = (void*)ws;
        ws += (bytes + 255) & ~((size_t)255);
        return p;
    };
    float* x_cur  = (float*)carve(sizeof(float) * (size_t)NNODES * EMB);  // 38.4 MB
    float* x_next = (float*)carve(sizeof(float) * (size_t)NNODES * EMB);  // 38.4 MB
    float* dinv   = (float*)carve(sizeof(float) * NNODES);                // 0.6 MB
    float* enorm  = (float*)carve(sizeof(float) * NEDGES);                // 9.6 MB

    const int vecN      = NNODES * 16;          // float4 elements = 2,400,000
    const int vecBlocks = vecN / 256;           // 9375 exactly
    const int edgBlocks = NEDGES / 256;         // 9375 exactly
    const int nodBlocks = (NNODES + 255) / 256; // 586

    // degrees + symmetric norms
    k_fill1    <<<nodBlocks, 256, 0, stream>>>(dinv);
    k_deg_accum<<<edgBlocks, 256, 0, stream>>>(col, dinv);
    k_dinv     <<<nodBlocks, 256, 0, stream>>>(dinv);
    k_edge_norm<<<edgBlocks, 256, 0, stream>>>(row, col, dinv, enorm);

    // x_cur = embeddings; final(out) = w0 * x
    k_init<<<vecBlocks, 256, 0, stream>>>((const float4*)user, (const float4*)item,
                                          lw, (float4*)x_cur, (float4*)out);

    for (int l = 0; l < NLAYERS; ++l) {
        k_selfloop<<<vecBlocks, 256, 0, stream>>>((const float4*)x_cur, dinv,
                                                  (float4*)x_next);
        k_scatter<<<SCATTER_BLOCKS, TILE, 0, stream>>>(row, col, enorm,
                                                       (const float4*)x_cur, x_next);
        k_axpy<<<vecBlocks, 256, 0, stream>>>((float4*)out, (const float4*)x_next,
                                              lw, l + 1);
        float* tmp = x_cur; x_cur = x_next; x_next = tmp;
    }

    k_l2norm<<<NNODES / 8, 256, 0, stream>>>(out);
}